// SAGEConvNet_5566277616451
// MI455X (gfx1250) — compile-verified
//
#include <hip/hip_runtime.h>

#define D 96

typedef float v2f __attribute__((ext_vector_type(2)));
typedef float v8f __attribute__((ext_vector_type(8)));

// ---------------------------------------------------------------- zero (f4)
__global__ void __launch_bounds__(256) k_zero(float4* __restrict__ p, int n4) {
  int i = blockIdx.x * blockDim.x + threadIdx.x;
  if (i < n4) p[i] = make_float4(0.f, 0.f, 0.f, 0.f);
}

// ------------------------------------------------------------- degree count
__global__ void __launch_bounds__(256) k_degree(const long long* __restrict__ dst,
                                                float* __restrict__ cnt, int E) {
  int e = blockIdx.x * blockDim.x + threadIdx.x;
  if (e < E) atomicAdd(cnt + (int)dst[e], 1.0f);
}

// ----------------------------------------------------- cnt -> 1/max(cnt,1)
__global__ void __launch_bounds__(256) k_inv(float* __restrict__ cnt, int n) {
  int i = blockIdx.x * blockDim.x + threadIdx.x;
  if (i < n) cnt[i] = 1.0f / fmaxf(cnt[i], 1.0f);
}

// ---------------------------------------------- edge scatter: agg[dst]+=x[src]
// one wave32 per edge; lane handles dims {lane, lane+32, lane+64}
__global__ void __launch_bounds__(256) k_scatter(const float* __restrict__ feat,
                                                 const long long* __restrict__ src,
                                                 const long long* __restrict__ dst,
                                                 float* __restrict__ agg, int E) {
  const int lane = threadIdx.x & 31;
  const int e = blockIdx.x * 8 + (threadIdx.x >> 5);
  if (e >= E) return;
  const int s = (int)src[e];
  const int d = (int)dst[e];
  const float* fp = feat + (size_t)s * D;
  float* ap = agg + (size_t)d * D;
#pragma unroll
  for (int i = 0; i < 3; ++i)
    atomicAdd(ap + lane + i * 32, fp[lane + i * 32]);
}

// --------------------------------------------------------- fused SAGE GEMM
// out[i,:] = act( (agg[i,:] * inv[i]) @ Wl^T + b + x[i,:] @ Wr^T )
// One wave32 per 16-row strip; 6 column tiles of 16; K in steps of 4 via
// V_WMMA_F32_16X16X4_F32 (exact fp32, matches reference precision).
// Weights are staged in LDS once per block, pre-swizzled into WMMA B-fragment
// order so the hot loop does conflict-free ds_load_b64 at lane*8 stride.
#define NFRAG (6 * 24 * 32)  // ct * kq * lane float2 fragments per matrix

__global__ void __launch_bounds__(128) k_sage_gemm(
    const float* __restrict__ agg, const float* __restrict__ xin,
    const float* __restrict__ inv, const float* __restrict__ Wl,
    const float* __restrict__ bias, const float* __restrict__ Wr,
    float* __restrict__ out, int n_nodes, int apply_elu) {
  __shared__ v2f fWl[NFRAG];  // 36 KB
  __shared__ v2f fWr[NFRAG];  // 36 KB

  // Cooperative fill: fragment (ct, kq, lane) = {W[col][k], W[col][k+1]}
  // with col = ct*16 + (lane&15), k = 4*kq + 2*(lane>>4).
  for (int fi = threadIdx.x; fi < NFRAG; fi += 128) {
    const int fl = fi & 31;
    const int kq = (fi >> 5) % 24;
    const int fc = fi / 768;
    const int n = fc * 16 + (fl & 15);
    const int k = 4 * kq + ((fl >> 4) << 1);
    fWl[fi] = *(const v2f*)(Wl + (size_t)n * D + k);
    fWr[fi] = *(const v2f*)(Wr + (size_t)n * D + k);
  }
  __syncthreads();

  const int lane = threadIdx.x & 31;
  const int wave = threadIdx.x >> 5;                 // 0..3
  const int rowbase = (blockIdx.x * 4 + wave) * 16;  // 16-row strip per wave
  if (rowbase >= n_nodes) return;                    // wave-uniform (EXEC all-1s)

  const int r = lane & 15;   // A: row-in-tile / C: col-in-tile
  const int hi = lane >> 4;  // half-wave selector
  const int koff = hi << 1;  // K sub-offset {0,2} for A fragments
  const int row = rowbase + r;
  const float s = inv[row];  // mean normalization, lane-uniform per A row

  const float* aggp = agg + (size_t)row * D;
  const float* xp = xin + (size_t)row * D;

  for (int ct = 0; ct < 6; ++ct) {
    const int colbase = ct * 16;
    const float bc = bias[colbase + r];
    v8f acc;
#pragma unroll
    for (int v = 0; v < 8; ++v) acc[v] = bc;  // C init = bias broadcast

    const v2f* bl = fWl + ct * 768 + lane;  // conflict-free lane*8B stride
    const v2f* br = fWr + ct * 768 + lane;

#pragma unroll
    for (int kq = 0; kq < 24; ++kq) {
      // GEMM 1: (agg * inv) @ Wl^T
      v2f a1 = *(const v2f*)(aggp + 4 * kq + koff);
      a1 = a1 * s;
      v2f b1 = bl[kq * 32];
      acc = __builtin_amdgcn_wmma_f32_16x16x4_f32(false, a1, false, b1,
                                                  (short)0, acc, false, false);
      // GEMM 2: x @ Wr^T
      v2f a2 = *(const v2f*)(xp + 4 * kq + koff);
      v2f b2 = br[kq * 32];
      acc = __builtin_amdgcn_wmma_f32_16x16x4_f32(false, a2, false, b2,
                                                  (short)0, acc, false, false);
    }

    // C/D layout: acc[v] -> row rowbase + v + 8*hi, col colbase + r
    float* op = out + (size_t)(rowbase + 8 * hi) * D + colbase + r;
#pragma unroll
    for (int v = 0; v < 8; ++v) {
      float val = acc[v];
      if (apply_elu) val = (val > 0.f) ? val : (__expf(val) - 1.f);
      op[(size_t)v * D] = val;
    }
  }
}

extern "C" void kernel_launch(void* const* d_in, const int* in_sizes, int n_in,
                              void* d_out, int out_size, void* d_ws, size_t ws_size,
                              hipStream_t stream) {
  const float* x = (const float*)d_in[0];
  const long long* ei = (const long long*)d_in[1];  // int64 [2, E]
  const float* W1l = (const float*)d_in[2];
  const float* b1 = (const float*)d_in[3];
  const float* W1r = (const float*)d_in[4];
  const float* W2l = (const float*)d_in[5];
  const float* b2 = (const float*)d_in[6];
  const float* W2r = (const float*)d_in[7];
  float* out = (float*)d_out;

  const int N = in_sizes[0] / D;  // 50000
  const int E = in_sizes[1] / 2;  // 800000
  const long long* src = ei;
  const long long* dst = ei + E;

  // workspace layout (floats): [cnt/inv: 64K pad][agg: N*96][h: N*96] ~ 39 MB
  float* f = (float*)d_ws;
  float* cnt = f;
  float* agg = f + (1 << 16);
  float* h = agg + (size_t)N * D;

  const int nzero1 = ((1 << 16) + N * D) / 4;  // cnt + agg region, in float4s
  const int nzero2 = (N * D) / 4;              // agg region only
  const int gblocks = (N + 63) / 64;           // 64 rows (4 waves) per block

  // Layer 1
  k_zero<<<(nzero1 + 255) / 256, 256, 0, stream>>>((float4*)f, nzero1);
  k_degree<<<(E + 255) / 256, 256, 0, stream>>>(dst, cnt, E);
  k_inv<<<(N + 255) / 256, 256, 0, stream>>>(cnt, N);
  k_scatter<<<(E + 7) / 8, 256, 0, stream>>>(x, src, dst, agg, E);
  k_sage_gemm<<<gblocks, 128, 0, stream>>>(agg, x, cnt, W1l, b1, W1r, h, N, 1);

  // Layer 2
  k_zero<<<(nzero2 + 255) / 256, 256, 0, stream>>>((float4*)agg, nzero2);
  k_scatter<<<(E + 7) / 8, 256, 0, stream>>>(h, src, dst, agg, E);
  k_sage_gemm<<<gblocks, 128, 0, stream>>>(agg, h, cnt, W2l, b2, W2r, out, N, 0);
}